// CandyNet_7808250544136
// MI455X (gfx1250) — compile-verified
//
#include <hip/hip_runtime.h>
#include <math.h>

// ---- CDNA5 WMMA types -------------------------------------------------------
typedef float v2f __attribute__((ext_vector_type(2)));
typedef float v8f __attribute__((ext_vector_type(8)));

#define IMG_H 512
#define IMG_W 512
#define NBATCH 16
#define RAWS 28   // raw tile LDS stride (floats): 28*4B = 112B => 16B-aligned chunks

// D = A(16x4) * B(4x16) + C, f32 WMMA (the only f32 matrix op on gfx1250)
__device__ __forceinline__ v8f wmma4(v2f a, v2f b, v8f c) {
    return __builtin_amdgcn_wmma_f32_16x16x4_f32(
        /*neg_a=*/false, a, /*neg_b=*/false, b,
        /*c_mod=*/(short)0, c, /*reuse_a=*/false, /*reuse_b=*/false);
}

// Banded Gaussian matrix entry: W[k][n] = g[k-n] when 0 <= k-n <= 4 else 0
__device__ __forceinline__ float bandw(int k, int n,
                                       float g0, float g1, float g2,
                                       float g3, float g4) {
    int d = k - n;
    float w = 0.0f;
    w = (d == 0) ? g0 : w;
    w = (d == 1) ? g1 : w;
    w = (d == 2) ? g2 : w;
    w = (d == 3) ? g3 : w;
    w = (d == 4) ? g4 : w;
    return w;
}

// Async global->LDS 16B transfer (CDNA5 async path, tracked by ASYNCcnt).
// lds_off: LDS byte offset (flat-LDS low 32 bits == LDS offset per aperture rules).
__device__ __forceinline__ void async_copy_b128(unsigned lds_off, const float* gptr) {
    asm volatile("global_load_async_to_lds_b128 %0, %1, off"
                 :: "v"(lds_off), "v"(gptr) : "memory");
}
__device__ __forceinline__ void wait_async0() {
    asm volatile("s_wait_asynccnt 0x0" ::: "memory");
}

// Fused Canny: gaussian (WMMA) -> sobel -> mag/orient -> directional NMS -> threshold
// One wave32 per 16x16 output tile; private LDS slab per wave => no barriers.
__launch_bounds__(128, 1)
__global__ void canny_fused(const float* __restrict__ img,
                            const float* __restrict__ gh,
                            float* __restrict__ out) {
    // Per-wave LDS slabs (4 waves/block)
    __shared__ float lds_raw[4][24 * RAWS]; // raw tile, halo 4  (24x24, stride 28)
    __shared__ float lds_hb [4][24 * 21];   // h-blur, rows halo 4, cols halo 2 (24x20)
    __shared__ float lds_bl [4][20 * 21];   // full blur, halo 2 (20x20)
    __shared__ float lds_mag[4][18 * 19];   // grad magnitude, halo 1 (18x18)
    __shared__ float lds_gx [4][16 * 17];   // sum gx over channels (16x16)
    __shared__ float lds_gy [4][16 * 17];   // sum gy over channels (16x16)

    const int lane = threadIdx.x & 31;
    const int wv   = threadIdx.x >> 5;

    float* raw = lds_raw[wv];
    float* hb  = lds_hb[wv];
    float* bl  = lds_bl[wv];
    float* mg  = lds_mag[wv];
    float* gxs = lds_gx[wv];
    float* gys = lds_gy[wv];

    // LDS byte offset of this wave's raw slab for async-to-LDS addressing
    const unsigned raw_base = (unsigned)(size_t)(void*)raw;

    // tile -> (image, row, col);  32x32 tiles per 512x512 image
    const int tile = blockIdx.x * 4 + wv;
    const int im   = tile >> 10;
    const int t    = tile & 1023;
    const int orow = (t >> 5) << 4;
    const int ocol = (t & 31) << 4;

    const float g0 = gh[0], g1 = gh[1], g2 = gh[2], g3 = gh[3], g4 = gh[4];

    const int m  = lane & 15;   // row (A) or column (B/C/D) index inside 16x16
    const int hi = lane >> 4;   // half-wave select

    for (int ch = 0; ch < 3; ++ch) {
        const float* src = img + (size_t)(im * 3 + ch) * (IMG_H * IMG_W);

        // ---- async-fill raw 24x24 tile (origin orow-4, ocol-4) --------------
        // 24 rows x 6 chunks of 4 floats; tile cols are 4-aligned and the image
        // width is a multiple of 4, so each 16B chunk is fully in or fully out.
        for (int i = lane; i < 24 * 6; i += 32) {
            int rr = i / 6, j = i - rr * 6;
            int r  = orow - 4 + rr;
            int cb = ocol - 4 + 4 * j;
            if (r >= 0 && r < IMG_H && cb >= 0 && cb < IMG_W) {
                async_copy_b128(raw_base + (unsigned)((rr * RAWS + 4 * j) * 4),
                                src + r * IMG_W + cb);
            } else {
                float* dp = raw + rr * RAWS + 4 * j;     // zero-pad chunk
                dp[0] = 0.0f; dp[1] = 0.0f; dp[2] = 0.0f; dp[3] = 0.0f;
            }
        }
        // hint the next channel's tile into cache (global_prefetch_b8)
        if (ch < 2)
            __builtin_prefetch(src + IMG_H * IMG_W + (orow * IMG_W + ocol), 0, 1);
        wait_async0();   // raw tile resident in LDS

        // ---- horizontal gaussian via WMMA: D = raw(16x20) x Band(20x16) -----
        // 4 overlapping 16x16 tiles cover the needed 24x20 h-blur region
        for (int tr = 0; tr < 2; ++tr) {
            for (int tc = 0; tc < 2; ++tc) {
                const int r0 = orow - 4 + 8 * tr;   // rows r0..r0+15
                const int c0 = ocol - 2 + 4 * tc;   // cols c0..c0+15
                v8f acc = {0.f, 0.f, 0.f, 0.f, 0.f, 0.f, 0.f, 0.f};
                for (int q = 0; q < 5; ++q) {
                    const int kg = 4 * q + 2 * hi;
                    // A[m][k] = raw[r0+m][c0-2+k]
                    const int ar = (r0 + m) - (orow - 4);
                    const int ac = (c0 - 2 + kg) - (ocol - 4);
                    v2f a;
                    a.x = raw[ar * RAWS + ac];
                    a.y = raw[ar * RAWS + ac + 1];
                    // B[k][n] = band(k, n), n == lane&15
                    v2f b;
                    b.x = bandw(kg,     m, g0, g1, g2, g3, g4);
                    b.y = bandw(kg + 1, m, g0, g1, g2, g3, g4);
                    acc = wmma4(a, b, acc);
                }
                // D[v + 8*hi][n] -> hb
                for (int v = 0; v < 8; ++v) {
                    const int dr_ = (r0 + v + 8 * hi) - (orow - 4);
                    const int dc_ = (c0 + m) - (ocol - 2);
                    hb[dr_ * 21 + dc_] = acc[v];
                }
            }
        }

        // ---- vertical gaussian via WMMA: D = Band(16x20) x hb(20x16) --------
        for (int tr = 0; tr < 2; ++tr) {
            for (int tc = 0; tc < 2; ++tc) {
                const int r0 = orow - 2 + 4 * tr;
                const int c0 = ocol - 2 + 4 * tc;
                v8f acc = {0.f, 0.f, 0.f, 0.f, 0.f, 0.f, 0.f, 0.f};
                for (int q = 0; q < 5; ++q) {
                    const int kg = 4 * q + 2 * hi;
                    // A[m][k] = band(k, m)
                    v2f a;
                    a.x = bandw(kg,     m, g0, g1, g2, g3, g4);
                    a.y = bandw(kg + 1, m, g0, g1, g2, g3, g4);
                    // B[k][n] = hb[r0-2+k][c0+n]
                    const int br = (r0 - 2 + kg) - (orow - 4);
                    const int bc = (c0 + m) - (ocol - 2);
                    v2f b;
                    b.x = hb[br * 21 + bc];
                    b.y = hb[(br + 1) * 21 + bc];
                    acc = wmma4(a, b, acc);
                }
                // store blurred; zero outside image (reference zero-pads blurred)
                for (int v = 0; v < 8; ++v) {
                    const int gr = r0 + v + 8 * hi;
                    const int gc = c0 + m;
                    float val = acc[v];
                    if (gr < 0 || gr >= IMG_H || gc < 0 || gc >= IMG_W) val = 0.0f;
                    bl[(gr - (orow - 2)) * 21 + (gc - (ocol - 2))] = val;
                }
            }
        }

        // ---- sobel + magnitude (18x18, halo 1), accumulate over channels ----
        for (int i = lane; i < 18 * 18; i += 32) {
            int rr = i / 18, cc = i - rr * 18;
            int r = orow - 1 + rr, c = ocol - 1 + cc;
            float gx = 0.0f, gy = 0.0f, mv = 0.0f;
            if (r >= 0 && r < IMG_H && c >= 0 && c < IMG_W) {
                const int br = r - (orow - 2), bc = c - (ocol - 2);
                float a00 = bl[(br - 1) * 21 + (bc - 1)];
                float a01 = bl[(br - 1) * 21 + bc];
                float a02 = bl[(br - 1) * 21 + (bc + 1)];
                float a10 = bl[br * 21 + (bc - 1)];
                float a12 = bl[br * 21 + (bc + 1)];
                float a20 = bl[(br + 1) * 21 + (bc - 1)];
                float a21 = bl[(br + 1) * 21 + bc];
                float a22 = bl[(br + 1) * 21 + (bc + 1)];
                gx = (a00 - a02) + 2.0f * (a10 - a12) + (a20 - a22);
                gy = (a00 - a20) + 2.0f * (a01 - a21) + (a02 - a22);
                mv = sqrtf(gx * gx + gy * gy);
            }
            if (ch == 0) mg[rr * 19 + cc] = mv;
            else         mg[rr * 19 + cc] += mv;
            if (rr >= 1 && rr <= 16 && cc >= 1 && cc <= 16) {
                const int xr = rr - 1, xc = cc - 1;
                if (ch == 0) { gxs[xr * 17 + xc]  = gx; gys[xr * 17 + xc]  = gy; }
                else         { gxs[xr * 17 + xc] += gx; gys[xr * 17 + xc] += gy; }
            }
        }
    }

    // ---- orientation, directional NMS, threshold ---------------------------
    const float RAD2DEG = 180.0f / 3.14159f;
    for (int i = lane; i < 256; i += 32) {
        const int rr = i >> 4, cc = i & 15;
        const int r = orow + rr, c = ocol + cc;
        const float gxv = gxs[rr * 17 + cc];
        const float gyv = gys[rr * 17 + cc];
        const float mc  = mg[(rr + 1) * 19 + (cc + 1)];
        float orient = atan2f(gyv, gxv) * RAD2DEG + 180.0f;
        float qf = rintf(orient * (1.0f / 45.0f));        // RNE, matches jnp.round
        int ip = ((int)qf) & 7;
        int in_ = (ip + 4) & 7;
        // packed (dr+1), (dc+1) nibble tables for directions 0..7
        int drp = (int)((0x00012221u >> (4 * ip)) & 0xFu) - 1;
        int dcp = (int)((0x21000122u >> (4 * ip)) & 0xFu) - 1;
        int drn = (int)((0x00012221u >> (4 * in_)) & 0xFu) - 1;
        int dcn = (int)((0x21000122u >> (4 * in_)) & 0xFu) - 1;
        float pos = mc - mg[(rr + 1 + drp) * 19 + (cc + 1 + dcp)];
        float neg = mc - mg[(rr + 1 + drn) * 19 + (cc + 1 + dcn)];
        float thin = (fminf(pos, neg) > 0.0f) ? mc : 0.0f;
        out[((size_t)im * IMG_H + r) * IMG_W + c] = (thin < 10.0f) ? 0.0f : thin;
    }
}

extern "C" void kernel_launch(void* const* d_in, const int* in_sizes, int n_in,
                              void* d_out, int out_size, void* d_ws, size_t ws_size,
                              hipStream_t stream) {
    (void)in_sizes; (void)n_in; (void)out_size; (void)d_ws; (void)ws_size;
    const float* img = (const float*)d_in[0];   // [16,3,512,512] f32
    const float* gh  = (const float*)d_in[1];   // gaussian taps (5 floats)
    float* out = (float*)d_out;                 // [16,1,512,512] f32
    // 16 images * 32*32 tiles, 4 wave-tiles per 128-thread block
    const int blocks = NBATCH * 32 * 32 / 4;
    canny_fused<<<blocks, 128, 0, stream>>>(img, gh, out);
}